// SEG_loss_13640816132401
// MI455X (gfx1250) — compile-verified
//
#include <hip/hip_runtime.h>
#include <hip/hip_bf16.h>

// Problem constants (match the reference).
#define B_ 4
#define P_ 24
#define G_ 24
#define H_ 512
#define W_ 512
#define NPAIR (B_ * P_)          // 96
#define CHUNKS 32
#define ROWS_PER_CHUNK (H_ / CHUNKS)  // 16
#define IOU_THRESH 0.5f
#define LOG_CLAMP -100.0f

typedef __attribute__((ext_vector_type(2))) float v2f;
typedef __attribute__((ext_vector_type(8))) float v8f;

#if defined(__has_builtin)
#if __has_builtin(__builtin_amdgcn_global_load_async_to_lds_b32)
#define HAVE_ASYNC_LDS 1
#endif
#if __has_builtin(__builtin_amdgcn_wmma_f32_16x16x4_f32)
#define HAVE_WMMA_F32 1
#endif
#endif

// ---------------------------------------------------------------------------
// Kernel 1: box matching. One thread per (b, p) pair.
// ---------------------------------------------------------------------------
__global__ void seg_match_kernel(const float* __restrict__ pred_boxes,
                                 const float* __restrict__ gt_boxes,
                                 const int* __restrict__ pred_valid,
                                 const int* __restrict__ gt_valid,
                                 int* __restrict__ match /* [NPAIR][6] */) {
  int t = threadIdx.x;
  if (t >= NPAIR) return;
  int b = t / P_;
  int p = t % P_;
  const float* pb = pred_boxes + (size_t)(b * P_ + p) * 4;
  float a0 = pb[0], a1 = pb[1], a2 = pb[2], a3 = pb[3];
  float areaA = (a2 - a0) * (a3 - a1);

  float best = -1e30f;
  int bestg = 0;
  for (int g = 0; g < G_; ++g) {
    const float* gb = gt_boxes + (size_t)(b * G_ + g) * 4;
    float b0 = gb[0], b1 = gb[1], b2 = gb[2], b3 = gb[3];
    float areaB = (b2 - b0) * (b3 - b1);
    float ih = fmaxf(fminf(a2, b2) - fmaxf(a0, b0), 0.0f);
    float iw = fmaxf(fminf(a3, b3) - fmaxf(a1, b1), 0.0f);
    float inter = ih * iw;
    float uni = areaA + areaB - inter;
    float iou = (uni == 0.0f) ? 0.0f : inter / uni;
    if (gt_valid[b * G_ + g] == 0) iou = -1.0f;   // masked-out GT
    if (iou > best) { best = iou; bestg = g; }    // first-max == jnp.argmax
  }
  int keep = (best >= IOU_THRESH) && (pred_valid[b * P_ + p] != 0);

  // jnp.round is round-half-to-even -> rintf (RNE default mode).
  int y1 = max(0, (int)rintf(a0 * (float)H_));
  int x1 = max(0, (int)rintf(a1 * (float)W_));
  int y2 = min((int)rintf(a2 * (float)H_), H_ - 1);
  int x2 = min((int)rintf(a3 * (float)W_), W_ - 1);

  int* m = match + t * 6;
  m[0] = bestg; m[1] = keep; m[2] = y1; m[3] = x1; m[4] = y2; m[5] = x2;
}

// ---------------------------------------------------------------------------
// WMMA-based reduction of 256 LDS floats, executed by wave 0 only.
// Treat the 256 values as a 16x16 matrix M; D = sum_c A_c(16x4) * Ones(4x16)
// accumulates row sums into every column of D; column N=0 lives in lane 0
// (M=0..7 in acc[0..7]) and lane 16 (M=8..15).
// ---------------------------------------------------------------------------
__device__ __forceinline__ float wave0_reduce256(const float* __restrict__ red,
                                                 int lane) {
#ifdef HAVE_WMMA_F32
  int m = lane & 15;
  int kb = (lane >> 4) << 1;  // lanes 0-15 hold K=0,1 ; lanes 16-31 hold K=2,3
  v2f ones = {1.0f, 1.0f};
  v8f acc = {0.f, 0.f, 0.f, 0.f, 0.f, 0.f, 0.f, 0.f};
#pragma unroll
  for (int c = 0; c < 4; ++c) {
    v2f a;
    a.x = red[m * 16 + c * 4 + kb];
    a.y = red[m * 16 + c * 4 + kb + 1];
    acc = __builtin_amdgcn_wmma_f32_16x16x4_f32(
        /*neg_a=*/false, a, /*neg_b=*/false, ones,
        /*c_mod=*/(short)0, acc, /*reuse_a=*/false, /*reuse_b=*/false);
  }
  float s = acc[0] + acc[1] + acc[2] + acc[3] + acc[4] + acc[5] + acc[6] + acc[7];
  return __shfl(s, 0, 32) + __shfl(s, 16, 32);
#else
  float v = 0.0f;
  for (int i = lane; i < 256; i += 32) v += red[i];
#pragma unroll
  for (int off = 16; off > 0; off >>= 1) v += __shfl_down(v, off, 32);
  return __shfl(v, 0, 32);
#endif
}

// ---------------------------------------------------------------------------
// Kernel 2: per-pixel streaming reduction. grid = (NPAIR, CHUNKS), 256 thr.
// ---------------------------------------------------------------------------
__global__ void __launch_bounds__(256)
seg_pixel_kernel(const float* __restrict__ pred_masks,
                 const float* __restrict__ gt_masks,
                 const int* __restrict__ match,
                 float* __restrict__ partials /* [NPAIR*CHUNKS][5] */) {
  const int pair = blockIdx.x;
  const int chunk = blockIdx.y;
  const int tid = threadIdx.x;
  const int c0 = chunk * ROWS_PER_CHUNK;
  float* out = partials + (size_t)(pair * CHUNKS + chunk) * 5;

  __shared__ int msh[8];
#ifdef HAVE_ASYNC_LDS
  if (tid < 6) {
    // Builtin expects a NON-const AS(1) pointer (per hipcc diagnostic); the
    // C-style cast strips const (safe: load-to-LDS only reads through it).
    __attribute__((address_space(1))) int* gsrc =
        (__attribute__((address_space(1))) int*)(match + pair * 6 + tid);
    __attribute__((address_space(3))) int* ldst =
        (__attribute__((address_space(3))) int*)(&msh[tid]);
    __builtin_amdgcn_global_load_async_to_lds_b32(gsrc, ldst, 0, 0);
  }
#if __has_builtin(__builtin_amdgcn_s_wait_asynccnt)
  __builtin_amdgcn_s_wait_asynccnt(0);
#else
  asm volatile("s_wait_asynccnt 0" ::: "memory");
#endif
  __syncthreads();
#else
  if (tid < 6) msh[tid] = match[pair * 6 + tid];
  __syncthreads();
#endif
  const int g = msh[0], keep = msh[1], y1 = msh[2], x1 = msh[3],
            y2 = msh[4], x2 = msh[5];

  // Skip blocks that cannot contribute (uniform across block -> safe return).
  if (!keep || (c0 + ROWS_PER_CHUNK - 1) < y1 || c0 > y2) {
    if (tid < 5) out[tid] = 0.0f;
    return;
  }

  const int b = pair / P_;
  const int p = pair % P_;
  const float* pm = pred_masks + (size_t)(b * P_ + p) * H_ * W_;
  const float* gm = gt_masks + (size_t)(b * G_ + g) * H_ * W_;

  const int subrow = tid >> 7;     // 0..1
  const int col4 = tid & 127;      // 0..127 -> cols [4*col4, 4*col4+3]
  const int col = col4 * 4;

  float cnt = 0.f, bce = 0.f, inter = 0.f, sp = 0.f, st = 0.f;
#pragma unroll
  for (int rr = 0; rr < ROWS_PER_CHUNK; rr += 2) {
    const int row = c0 + rr + subrow;
    if (row < y1 || row > y2) continue;
    const float4 pv = *(reinterpret_cast<const float4*>(pm + (size_t)row * W_) + col4);
    const float4 tv = *(reinterpret_cast<const float4*>(gm + (size_t)row * W_) + col4);
    if (row + 2 <= y2) {  // speculative prefetch of next row pair
      __builtin_prefetch(pm + (size_t)(row + 2) * W_ + col, 0, 3);
      __builtin_prefetch(gm + (size_t)(row + 2) * W_ + col, 0, 3);
    }
    const float pa[4] = {pv.x, pv.y, pv.z, pv.w};
    const float ta[4] = {tv.x, tv.y, tv.z, tv.w};
#pragma unroll
    for (int j = 0; j < 4; ++j) {
      const int c = col + j;
      if (c < x1 || c > x2) continue;
      const float pp = pa[j];
      const float tt = ta[j];
      const float lp = fmaxf(logf(pp), LOG_CLAMP);
      const float l1p = fmaxf(log1pf(-pp), LOG_CLAMP);
      bce += -(tt * lp + (1.0f - tt) * l1p);
      cnt += 1.0f;
      sp += pp;
      st += tt;
      inter += pp * tt;
    }
  }

  __shared__ float red[5][256];
  red[0][tid] = cnt;
  red[1][tid] = bce;
  red[2][tid] = inter;
  red[3][tid] = sp;
  red[4][tid] = st;
  __syncthreads();

  if (tid < 32) {  // wave 0: WMMA row-sum reduction, EXEC all-1s in-wave
    const float r0 = wave0_reduce256(&red[0][0], tid);
    const float r1 = wave0_reduce256(&red[1][0], tid);
    const float r2 = wave0_reduce256(&red[2][0], tid);
    const float r3 = wave0_reduce256(&red[3][0], tid);
    const float r4 = wave0_reduce256(&red[4][0], tid);
    if (tid == 0) {
      out[0] = r0; out[1] = r1; out[2] = r2; out[3] = r3; out[4] = r4;
    }
  }
}

// ---------------------------------------------------------------------------
// Kernel 3: combine partials -> final scalar. One block of 128 threads.
// ---------------------------------------------------------------------------
__global__ void seg_final_kernel(const float* __restrict__ partials,
                                 const int* __restrict__ match,
                                 float* __restrict__ out) {
  __shared__ float contrib[128];
  __shared__ float keeps[128];
  const int t = threadIdx.x;
  float c = 0.0f, k = 0.0f;
  if (t < NPAIR) {
    float cnt = 0.f, bces = 0.f, inter = 0.f, sp = 0.f, st = 0.f;
    for (int ch = 0; ch < CHUNKS; ++ch) {
      const float* q = partials + (size_t)(t * CHUNKS + ch) * 5;
      cnt += q[0]; bces += q[1]; inter += q[2]; sp += q[3]; st += q[4];
    }
    const int keep = match[t * 6 + 1];
    const float bce = bces / fmaxf(cnt, 1.0f);
    const float dice = 1.0f - (2.0f * inter + 1.0f) / (sp + st + 1.0f);
    if (keep) { c = bce + dice; k = 1.0f; }
  }
  contrib[t] = c;
  keeps[t] = k;
  __syncthreads();
  if (t == 0) {
    float tot = 0.0f, n = 0.0f;
    for (int i = 0; i < 128; ++i) { tot += contrib[i]; n += keeps[i]; }
    out[0] = tot / fmaxf(n, 1.0f);
  }
}

// ---------------------------------------------------------------------------
// Launch wrapper.
// ---------------------------------------------------------------------------
extern "C" void kernel_launch(void* const* d_in, const int* in_sizes, int n_in,
                              void* d_out, int out_size, void* d_ws,
                              size_t ws_size, hipStream_t stream) {
  (void)in_sizes; (void)n_in; (void)out_size; (void)ws_size;
  const float* pred_masks = (const float*)d_in[0];
  const float* pred_boxes = (const float*)d_in[1];
  const float* gt_boxes   = (const float*)d_in[2];
  const float* gt_masks   = (const float*)d_in[3];
  const int*   pred_valid = (const int*)d_in[4];
  const int*   gt_valid   = (const int*)d_in[5];

  int* match = (int*)d_ws;                                   // 96*6 ints
  float* partials = (float*)((char*)d_ws + 4096);            // 96*32*5 floats

  seg_match_kernel<<<1, 128, 0, stream>>>(pred_boxes, gt_boxes, pred_valid,
                                          gt_valid, match);
  dim3 grid(NPAIR, CHUNKS);
  seg_pixel_kernel<<<grid, 256, 0, stream>>>(pred_masks, gt_masks, match,
                                             partials);
  seg_final_kernel<<<1, 128, 0, stream>>>(partials, match, (float*)d_out);
}